// DCELoss_26182120636838
// MI455X (gfx1250) — compile-verified
//
#include <hip/hip_runtime.h>
#include <math.h>

// ---------------------------------------------------------------------------
// DCE loss for MI455X (gfx1250).
// Heavy part: G = X @ X^T, X = [pred; hr] : 64 x 884736 fp32 (Gram of stacked
// rows gives P.P^T, P.H^T, H.H^T in one pass). Memory-bound: 226 MB @
// 23.3 TB/s ~ 9.7 us floor; AI ~16 MAC/byte, fp32 WMMA (V_WMMA_F32_16X16X4)
// has ample headroom. Data path: async global->LDS (ASYNCcnt) with a double-
// buffered LDS tile so the DMA of chunk n+1 overlaps WMMA on chunk n and no
// byte crosses the VGPR file on the way in.
// ---------------------------------------------------------------------------

typedef float v2f __attribute__((ext_vector_type(2)));
typedef float v8f __attribute__((ext_vector_type(8)));

#define KTOT   (96 * 96 * 96)   // 884736 elements per row
#define BROWS  32               // batch
#define XROWS  64               // stacked [pred; hr]
#define KC     96               // K-chunk per LDS buffer (KTOT % KC == 0)
#define LDSS   100              // padded row stride in floats (400 B):
                                // 16B-aligned rows, banks rotate 36r mod 64
#define F4PR   (KC / 4)         // 24 x b128 per row
#define NLD    ((XROWS * F4PR) / 256)  // 6 async b128 per thread per chunk
#define BUFB   (XROWS * LDSS * 4)      // 25600 B per buffer
#define NWG    512
#define NTHR   256              // 8 wave32 waves

__global__ __launch_bounds__(NTHR)
void dce_gram_kernel(const float* __restrict__ pred,
                     const float* __restrict__ hr,
                     float* __restrict__ G /* 64x64 accumulator */) {
    __shared__ float tile[2][XROWS * LDSS];

    const int tid  = threadIdx.x;
    const int lane = tid & 31;
    const int wave = tid >> 5;

    // Fragment addressing (ISA 7.12.2, 32-bit A 16x4):
    //   lane L, VGPR v -> m = L%16, k = (L/16)*2 + v.  B mirrored (n = L%16).
    const int m    = lane & 15;
    const int koff = (lane >> 4) * 2;

    // 64x64 output = 4x4 grid of 16x16 tiles; wave w owns (ti,tj0),(ti,tj0+1)
    // which share the A fragment.
    const int ti  = wave >> 1;
    const int tj0 = (wave & 1) * 2;

    // Per-thread async-copy slots: fixed (row, 16B-column) pairs; per chunk
    // only the K-base offset changes. 6 x b128 per thread covers 64 x KC.
    uint64_t gbase[NLD];
    unsigned lds0[NLD];
    const unsigned ldsbase = (unsigned)(uintptr_t)(&tile[0][0]);  // addr[31:0]
    #pragma unroll
    for (int s = 0; s < NLD; ++s) {
        const int idx = s * NTHR + tid;
        const int r   = idx / F4PR;
        const int f   = idx % F4PR;
        const float* row = (r < BROWS) ? (pred + (size_t)r * KTOT)
                                       : (hr + (size_t)(r - BROWS) * KTOT);
        gbase[s] = (uint64_t)(uintptr_t)row + (uint64_t)(f * 16);
        lds0[s]  = ldsbase + (unsigned)(r * LDSS * 4 + f * 16);
    }

    v8f acc0 = {};
    v8f acc1 = {};

    const int nchunks = KTOT / KC;  // 9216

    // Prologue: fill buffer 0 with this block's first chunk.
    {
        const uint64_t kb = (uint64_t)blockIdx.x * (KC * 4);
        #pragma unroll
        for (int s = 0; s < NLD; ++s) {
            const uint64_t ga = gbase[s] + kb;
            asm volatile("global_load_async_to_lds_b128 %0, %1, off"
                         :: "v"(lds0[s]), "v"(ga) : "memory");
        }
    }

    int t = 0;
    for (int cc = blockIdx.x; cc < nchunks; cc += gridDim.x, ++t) {
        const int nc = cc + (int)gridDim.x;  // next chunk (uniform per block)
        if (nc < nchunks) {
            // Kick off the next chunk into the other buffer, then wait for
            // only the previous 6 fills (in-order completion) to land.
            const unsigned poff = (unsigned)((t + 1) & 1) * BUFB;
            const uint64_t kb   = (uint64_t)nc * (KC * 4);
            #pragma unroll
            for (int s = 0; s < NLD; ++s) {
                const uint64_t ga = gbase[s] + kb;
                const unsigned la = lds0[s] + poff;
                asm volatile("global_load_async_to_lds_b128 %0, %1, off"
                             :: "v"(la), "v"(ga) : "memory");
            }
            asm volatile("s_wait_asynccnt 0x6" ::: "memory");
        } else {
            asm volatile("s_wait_asynccnt 0x0" ::: "memory");
        }
        __syncthreads();  // all waves' fills of buf[t&1] visible

        const float* buf = &tile[t & 1][0];
        // 24 WMMA k-steps per chunk; all lanes uniform -> EXEC all ones.
        #pragma unroll 4
        for (int kb = 0; kb < KC; kb += 4) {
            const float* pa  = &buf[(ti * 16 + m) * LDSS + kb + koff];
            const float* pb0 = &buf[(tj0 * 16 + m) * LDSS + kb + koff];
            const float* pb1 = &buf[((tj0 + 1) * 16 + m) * LDSS + kb + koff];
            v2f a, b0, b1;
            a.x  = pa[0];  a.y  = pa[1];
            b0.x = pb0[0]; b0.y = pb0[1];
            b1.x = pb1[0]; b1.y = pb1[1];
            acc0 = __builtin_amdgcn_wmma_f32_16x16x4_f32(
                false, a, false, b0, (short)0, acc0, false, false);
            acc1 = __builtin_amdgcn_wmma_f32_16x16x4_f32(
                false, a, false, b1, (short)0, acc1, false, false);
        }
        __syncthreads();  // compute done before buf[t&1] is refilled at t+2
    }

    // C/D layout: VGPR r -> M = r (lanes 0-15) or r+8 (lanes 16-31), N = lane%16.
    const int rowbase = ti * 16 + ((lane >> 4) ? 8 : 0);
    #pragma unroll
    for (int r = 0; r < 8; ++r) {
        atomicAdd(&G[(rowbase + r) * XROWS + tj0 * 16 + m],       acc0[r]);
        atomicAdd(&G[(rowbase + r) * XROWS + (tj0 + 1) * 16 + m], acc1[r]);
    }
}

// Single wave32: lane i handles batch row i, shuffle-reduce the loss.
__global__ __launch_bounds__(32)
void dce_finalize_kernel(const float* __restrict__ G, float* __restrict__ out) {
    const int i = threadIdx.x;  // 0..31

    const float pn_i  = sqrtf(G[i * XROWS + i]);
    const float hsq_i = G[(BROWS + i) * XROWS + (BROWS + i)];

    float Qp = 0.0f, Qn = 0.0f;
    for (int j = 0; j < BROWS; ++j) {
        const float pn_j  = sqrtf(G[j * XROWS + j]);
        const float hsq_j = G[(BROWS + j) * XROWS + (BROWS + j)];
        const float hn_j  = sqrtf(hsq_j);

        const float s_srsr = G[i * XROWS + j] / (pn_i * pn_j);
        const float s_srhr = G[i * XROWS + BROWS + j] / (pn_i * hn_j);
        const float ghh    = G[(BROWS + i) * XROWS + BROWS + j];

        const float d2   = fmaxf(hsq_i + hsq_j - 2.0f * ghh, 0.0f);
        const float dist = sqrtf(d2);
        // dist==0: log10 -> -inf, -20*(-inf)=+inf, min(+inf,0)=0 -> mask false.
        const float M    = fminf(-20.0f * log10f(dist), 0.0f);
        const bool  mask = fabsf(M) > 30.0f;

        const float w = (expf(s_srsr) + 2.0f * expf(s_srhr)) * 2.0f;  // /T=0.5
        if (mask) Qp += w; else Qn += w;
    }

    float t = logf(Qp / Qn);
    #pragma unroll
    for (int off = 16; off > 0; off >>= 1) t += __shfl_down(t, off, 32);
    if (i == 0) out[0] = (-1.0f / (float)BROWS) * t;
}

extern "C" void kernel_launch(void* const* d_in, const int* in_sizes, int n_in,
                              void* d_out, int out_size, void* d_ws, size_t ws_size,
                              hipStream_t stream) {
    const float* pred = (const float*)d_in[0];
    const float* hr   = (const float*)d_in[1];
    float* G = (float*)d_ws;  // 64x64 fp32 Gram accumulator (16 KB)

    hipMemsetAsync(G, 0, XROWS * XROWS * sizeof(float), stream);
    dce_gram_kernel<<<NWG, NTHR, 0, stream>>>(pred, hr, G);
    dce_finalize_kernel<<<1, 32, 0, stream>>>(G, (float*)d_out);
}